// CoverageLoss_55276229099893
// MI455X (gfx1250) — compile-verified
//
#include <hip/hip_runtime.h>

typedef __attribute__((ext_vector_type(2))) float v2f;
typedef __attribute__((ext_vector_type(8))) float v8f;

namespace {
constexpr int kN     = 32;             // images
constexpr int kBp    = 8;              // boxes per image
constexpr int kF     = 16;
constexpr int kFP    = 64;
constexpr int kPPI   = kF * kFP;       // 1024 fragment points per image
constexpr int kNB    = 100;            // boundary samples
constexpr int kTiles = 7;              // ceil(100/16) -> padded to 112
constexpr int kWaves = kN * kPPI / 16; // 2048 waves, 16 points each
}

// One wave handles 16 fragment points of one image against all 8 boxes.
// Distance block via V_WMMA_F32_16X16X4_F32 using
//   ||p-q||^2 = ||p||^2 + ( A x B ),  A[m]=[-2px,-2py,1,0], B[:,n]=[qx,qy,||q||^2,0]^T
__global__ __launch_bounds__(128) void coverage_main(
    const float* __restrict__ boxes,   // (N*Bp, 4) center format [xc,yc,w,h]
    const float* __restrict__ frags,   // (N, F, FP, 2)
    float* __restrict__ partials)      // (kWaves)
{
  const int  lane = threadIdx.x & 31;
  const int  wave = (int)((blockIdx.x * blockDim.x + threadIdx.x) >> 5);
  const int  img  = wave >> 6;   // 64 waves per image
  const int  grp  = wave & 63;   // 16-point group within image
  const int  m    = lane & 15;
  const bool hiH  = lane >= 16;

  // ---- my fragment point (lane m owns point m of the group) ----
  const float2* fp = reinterpret_cast<const float2*>(frags) + img * kPPI + grp * 16;
  const float2 p = fp[m];

  // ---- A matrix (16x4 f32): VGPR0 = K0 / K2(halves), VGPR1 = K1 / K3 ----
  v2f A;
  A.x = hiH ? 1.0f : (-2.0f * p.x);   // K=0 : -2px   | K=2 : 1
  A.y = hiH ? 0.0f : (-2.0f * p.y);   // K=1 : -2py   | K=3 : 0

  // ---- per-lane data for my output rows r = j + (hiH ? 8 : 0) ----
  float rpx[8], rpy[8], rpn2[8];
  #pragma unroll
  for (int j = 0; j < 8; ++j) {
    const int src = j + (hiH ? 8 : 0);
    rpx[j]  = __shfl(p.x, src, 32);
    rpy[j]  = __shfl(p.y, src, 32);
    rpn2[j] = rpx[j] * rpx[j] + rpy[j] * rpy[j];
  }

  // ---- unit-square boundary sample s = 16*t + m (STEP=25 -> 100 samples) ----
  constexpr float inv24 = 1.0f / 24.0f;
  float bx[kTiles], by[kTiles], pad[kTiles];
  #pragma unroll
  for (int t = 0; t < kTiles; ++t) {
    const int s = t * 16 + m;
    float x, y;
    if (s < 50) {                      // left/right edges: x in {0,1}, y = j/24
      x = (s >= 25) ? 1.0f : 0.0f;
      y = (float)(s % 25) * inv24;
    } else {                           // bottom/top: x = i/24, y in {0,1}
      const int u = s - 50;
      x = (float)(u >> 1) * inv24;
      y = (float)(u & 1);
    }
    const bool ok = s < kNB;
    bx[t]  = ok ? x : 0.0f;
    by[t]  = ok ? y : 0.0f;
    pad[t] = ok ? 0.0f : 3.0e37f;      // sentinel so padded columns never win the min
  }

  const float4* bptr = reinterpret_cast<const float4*>(boxes) + img * kBp;

  float best[8];
  #pragma unroll
  for (int j = 0; j < 8; ++j) best[j] = 3.0e38f;

  #pragma unroll
  for (int b = 0; b < kBp; ++b) {
    const float4 cb = bptr[b];
    const float lox = cb.x - 0.5f * cb.z, hix = cb.x + 0.5f * cb.z;
    const float loy = cb.y - 0.5f * cb.w, hiy = cb.y + 0.5f * cb.w;

    v8f acc;
    #pragma unroll
    for (int t = 0; t < kTiles; ++t) {
      const float qx  = bx[t] * cb.z + lox;        // boundary * wh + lo
      const float qy  = by[t] * cb.w + loy;
      const float qn2 = qx * qx + qy * qy + pad[t];
      v2f B;
      B.x = hiH ? qn2  : qx;                       // K=0 : qx | K=2 : ||q||^2
      B.y = hiH ? 0.0f : qy;                       // K=1 : qy | K=3 : 0
      v8f c = {};
      v8f d = __builtin_amdgcn_wmma_f32_16x16x4_f32(
          /*neg_a=*/false, A, /*neg_b=*/false, B,
          /*c_mod=*/(short)0, c, /*reuse_a=*/false, /*reuse_b=*/false);
      if (t == 0) {
        acc = d;
      } else {
        #pragma unroll
        for (int j = 0; j < 8; ++j) acc[j] = fminf(acc[j], d[j]);
      }
    }

    // Branchless inside mask (reference: all([p-lo, hi-p] >= 0)) + ||p||^2 fold.
    // min3/min tree -> v_min3_num_f32; one v_cmp + v_cndmask per row, no exec games.
    #pragma unroll
    for (int j = 0; j < 8; ++j) {
      const float mn = fminf(fminf(fminf(rpx[j] - lox, hix - rpx[j]),
                                   rpy[j] - loy),
                             hiy - rpy[j]);
      const float outside_val = rpn2[j] + acc[j];
      const float v = (mn >= 0.0f) ? 0.0f : outside_val;
      best[j] = fminf(best[j], v);
    }
  }

  // one cross-lane min per row (within each 16-lane half), then sum rows
  float s = 0.0f;
  #pragma unroll
  for (int j = 0; j < 8; ++j) {
    float v = best[j];
    v = fminf(v, __shfl_xor(v, 1, 32));
    v = fminf(v, __shfl_xor(v, 2, 32));
    v = fminf(v, __shfl_xor(v, 4, 32));
    v = fminf(v, __shfl_xor(v, 8, 32));
    s += v;
  }
  const float tot = __shfl(s, 0, 32) + __shfl(s, 16, 32);
  if (lane == 0) partials[wave] = tot;
}

__global__ __launch_bounds__(256) void coverage_reduce(
    const float* __restrict__ partials, float* __restrict__ out,
    int n, float scale)
{
  __shared__ float sm[256];
  float s = 0.0f;
  for (int i = threadIdx.x; i < n; i += 256) s += partials[i];
  sm[threadIdx.x] = s;
  __syncthreads();
  #pragma unroll
  for (int off = 128; off > 0; off >>= 1) {
    if ((int)threadIdx.x < off) sm[threadIdx.x] += sm[threadIdx.x + off];
    __syncthreads();
  }
  if (threadIdx.x == 0) out[0] = sm[0] * scale;
}

extern "C" void kernel_launch(void* const* d_in, const int* in_sizes, int n_in,
                              void* d_out, int out_size, void* d_ws, size_t ws_size,
                              hipStream_t stream) {
  const float* boxes = (const float*)d_in[0];   // (256, 4) f32
  const float* frags = (const float*)d_in[1];   // (32, 16, 64, 2) f32
  // d_in[2] = obj_to_img: unused by the reference math
  float* partials = (float*)d_ws;               // kWaves floats = 8 KB
  float* out      = (float*)d_out;              // 1 float

  coverage_main<<<dim3(kWaves / 4), dim3(128), 0, stream>>>(boxes, frags, partials);
  coverage_reduce<<<dim3(1), dim3(256), 0, stream>>>(partials, out, kWaves,
                                                     1.0f / (float)(kFP * kN));
}